// CausalLinearAttention_66391604462233
// MI455X (gfx1250) — compile-verified
//
#include <hip/hip_runtime.h>
#include <cmath>

typedef __attribute__((ext_vector_type(2))) float v2f;
typedef __attribute__((ext_vector_type(4))) float v4f;
typedef __attribute__((ext_vector_type(8))) float v8f;

// D = A(16x4,f32) * B(4x16,f32) + C(16x16,f32)
#define WMMA4(a, b, c) __builtin_amdgcn_wmma_f32_16x16x4_f32( \
    false, (a), false, (b), (short)0, (c), false, false)

namespace {
constexpr int kB = 8, kL = 512, kN = 64, kF = 128, kH = 8, kC = 16;
constexpr int kChunks = kL / kC;   // 32 chunks of 16 timesteps
constexpr int XS = 132;            // padded LDS stride for the 16x128 x tile
constexpr int ST = 18;             // padded LDS stride for 16x16 tiles
constexpr float kEps = 1e-6f;
}

// phi(x) = elu(x) + 1, branchless: x>0 -> x+1, x<=0 -> exp(x)
__device__ __forceinline__ float phi(float x) {
  return fmaxf(x, 0.0f) + __expf(fminf(x, 0.0f));
}

__global__ __launch_bounds__(256, 1)
void cla_fused_kernel(const float* __restrict__ qin,
                      const float* __restrict__ Wq, const float* __restrict__ bq,
                      const float* __restrict__ Wk, const float* __restrict__ bk,
                      const float* __restrict__ Wv, const float* __restrict__ bv,
                      float* __restrict__ out)
{
  __shared__ float xs[kC * XS];          // current 16x128 input tile
  __shared__ float shQ[kH][kC * ST];     // per-head phi(Q) tile
  __shared__ float shK[kH][kC * ST];     // per-head phi(K) tile
  __shared__ float shV[kH][kC * ST];     // per-head V tile
  __shared__ float shA[kH][kC * ST];     // per-head masked Q.K^T tile
  __shared__ float shS[kH][kC * ST];     // per-head running KV state (E x Ev)
  __shared__ float shKcum[kH][16];       // per-head running sum of phi(K)
  __shared__ float shZ[kH][16];          // per-head per-row normalizer
  __shared__ float ostage[kC * kF];      // 16 x 128 output staging

  const int tid   = threadIdx.x;
  const int lane  = tid & 31;
  const int w     = tid >> 5;            // wave index == head index
  const int col   = lane & 15;           // tile column / A-frag row
  const int hi    = lane >> 4;           // lane half (0/1)
  const int khalf = hi << 1;             // k-pair base within 4-wide block
  const int mrow  = hi << 3;             // C/D layout row base (0/8)

  const int g = blockIdx.x;              // sequence id in [0, B*N)
  const int b = g >> 6;
  const int n = g & 63;

  float* Qt = shQ[w];
  float* Kt = shK[w];
  float* Vt = shV[w];
  float* At = shA[w];
  float* St = shS[w];

  // zero wave-private running state
  for (int i = lane; i < kC * ST; i += 32) St[i] = 0.0f;
  if (lane < 16) shKcum[w][lane] = 0.0f;

  const int wcol = w * 16 + col;         // this lane's output column in W
  const float bqv = bq[wcol];
  const float bkv = bk[wcol];
  const float bvv = bv[wcol];

  const int cr = tid >> 4;               // cooperative row 0..15
  const int cc = (tid & 15) << 3;        // cooperative col 0..120 (step 8)

  for (int t = 0; t < kChunks; ++t) {
    // ---- cooperative load of x chunk: 16 timesteps x 128 features ----
    {
      const float* src =
          qin + (((size_t)(b * kL + t * kC + cr) * kN + n) * kF + cc);
      v4f a0 = *reinterpret_cast<const v4f*>(src);
      v4f a1 = *reinterpret_cast<const v4f*>(src + 4);
      *reinterpret_cast<v4f*>(&xs[cr * XS + cc]) = a0;
      *reinterpret_cast<v4f*>(&xs[cr * XS + cc + 4]) = a1;
      if (t + 1 < kChunks) {             // speculative prefetch of next chunk
        __builtin_prefetch(src + (size_t)kC * kN * kF, 0, 0);
      }
    }
    __syncthreads();

    // ---- per-head projections: (16x128) @ (128x16), 32 WMMA k-steps each ----
    v8f cq = {}; v8f ck = {}; v8f cv = {};
    for (int s = 0; s < 32; ++s) {
      const int k0 = 4 * s + khalf;
      v2f a = *reinterpret_cast<v2f*>(&xs[col * XS + k0]);
      v2f wb;
      wb.x = Wq[k0 * kF + wcol]; wb.y = Wq[(k0 + 1) * kF + wcol];
      cq = WMMA4(a, wb, cq);
      wb.x = Wk[k0 * kF + wcol]; wb.y = Wk[(k0 + 1) * kF + wcol];
      ck = WMMA4(a, wb, ck);
      wb.x = Wv[k0 * kF + wcol]; wb.y = Wv[(k0 + 1) * kF + wcol];
      cv = WMMA4(a, wb, cv);
    }

    // bias + feature map phi(x)=elu(x)+1, stash tiles to LDS (C-layout)
    for (int r = 0; r < 8; ++r) {
      const int m = r + mrow;
      Qt[m * ST + col] = phi(cq[r] + bqv);
      Kt[m * ST + col] = phi(ck[r] + bkv);
      Vt[m * ST + col] = cv[r] + bvv;
    }
    asm volatile("" ::: "memory");

    // ---- A = tril(Qt @ Kt^T), inclusive causal mask ----
    v8f aat = {};
    for (int s = 0; s < 4; ++s) {
      const int k0 = 4 * s + khalf;
      v2f a  = *reinterpret_cast<v2f*>(&Qt[col * ST + k0]);
      v2f bb = *reinterpret_cast<v2f*>(&Kt[col * ST + k0]);
      aat = WMMA4(a, bb, aat);
    }
    for (int r = 0; r < 8; ++r) {
      const int m = r + mrow;
      At[m * ST + col] = (col <= m) ? aat[r] : 0.0f;
    }
    asm volatile("" ::: "memory");

    // ---- normalizer: Z = 1 / (rowsum(A) + Qt . kcum_prev + eps) ----
    if (lane < 16) {
      float den = 0.0f;
      for (int c2 = 0; c2 < 16; ++c2) den += At[lane * ST + c2];
      float dq = 0.0f;
      for (int e = 0; e < 16; ++e) dq += Qt[lane * ST + e] * shKcum[w][e];
      shZ[w][lane] = __builtin_amdgcn_rcpf(den + dq + kEps);
    }
    asm volatile("" ::: "memory");

    // ---- O = A @ Vt + Qt @ S_prev ----
    v8f oac = {};
    for (int s = 0; s < 4; ++s) {
      const int k0 = 4 * s + khalf;
      v2f a = *reinterpret_cast<v2f*>(&At[col * ST + k0]);
      v2f bb; bb.x = Vt[k0 * ST + col]; bb.y = Vt[(k0 + 1) * ST + col];
      oac = WMMA4(a, bb, oac);
    }
    for (int s = 0; s < 4; ++s) {
      const int k0 = 4 * s + khalf;
      v2f a = *reinterpret_cast<v2f*>(&Qt[col * ST + k0]);
      v2f bb; bb.x = St[k0 * ST + col]; bb.y = St[(k0 + 1) * ST + col];
      oac = WMMA4(a, bb, oac);
    }

    // ---- S += Kt^T @ Vt (accumulate in C-layout registers, write back) ----
    v8f sac;
    for (int r = 0; r < 8; ++r) sac[r] = St[(r + mrow) * ST + col];
    for (int s = 0; s < 4; ++s) {
      const int k0 = 4 * s + khalf;
      v2f a;  a.x  = Kt[k0 * ST + col]; a.y  = Kt[(k0 + 1) * ST + col];
      v2f bb; bb.x = Vt[k0 * ST + col]; bb.y = Vt[(k0 + 1) * ST + col];
      sac = WMMA4(a, bb, sac);
    }
    for (int r = 0; r < 8; ++r) St[(r + mrow) * ST + col] = sac[r];

    // ---- kcum += colsum(Kt) ----
    if (lane < 16) {
      float s = 0.0f;
      for (int c2 = 0; c2 < 16; ++c2) s += Kt[c2 * ST + lane];
      shKcum[w][lane] += s;
    }
    asm volatile("" ::: "memory");

    // ---- scale by Z, stage into shared output tile ----
    for (int r = 0; r < 8; ++r) {
      const int m = r + mrow;
      ostage[m * kF + w * 16 + col] = oac[r] * shZ[w][m];
    }
    __syncthreads();

    // ---- cooperative coalesced store of 16 x 128 output rows ----
    {
      float* dst =
          out + (((size_t)(b * kL + t * kC + cr) * kN + n) * kF + cc);
      *reinterpret_cast<v4f*>(dst)     = *reinterpret_cast<v4f*>(&ostage[cr * kF + cc]);
      *reinterpret_cast<v4f*>(dst + 4) = *reinterpret_cast<v4f*>(&ostage[cr * kF + cc + 4]);
    }
  }
}

extern "C" void kernel_launch(void* const* d_in, const int* in_sizes, int n_in,
                              void* d_out, int out_size, void* d_ws, size_t ws_size,
                              hipStream_t stream) {
  (void)in_sizes; (void)n_in; (void)out_size; (void)d_ws; (void)ws_size;
  const float* q  = (const float*)d_in[0];
  const float* Wq = (const float*)d_in[1];
  const float* bq = (const float*)d_in[2];
  const float* Wk = (const float*)d_in[3];
  const float* bk = (const float*)d_in[4];
  const float* Wv = (const float*)d_in[5];
  const float* bv = (const float*)d_in[6];
  cla_fused_kernel<<<dim3(kB * kN), dim3(256), 0, stream>>>(
      q, Wq, bq, Wk, bk, Wv, bv, (float*)d_out);
}